// RadixAttention_13408887898621
// MI455X (gfx1250) — compile-verified
//
#include <hip/hip_runtime.h>
#include <math.h>

// Problem constants (match reference)
#define B_    2
#define E_    1024
#define P_    1024
#define HQ_   32
#define HKV_  8
#define D_    128
#define G_    4
#define SCALE_ 0.08838834764831845f   // 1/sqrt(128)

typedef float v2f __attribute__((ext_vector_type(2)));
typedef float v8f __attribute__((ext_vector_type(8)));

// LDS row stride (floats) for K and V tiles: multiple of 4 (16B-aligned rows for
// async b128 writes); 140*l mod 64 distinct for l=0..15 -> conflict-free reads.
#define RSTR 140
#define SSTR 18    // per-wave P staging [16][16], rows 8B aligned

__device__ __forceinline__ v8f wmma4_f32(v2f a, v2f b, v8f c) {
  return __builtin_amdgcn_wmma_f32_16x16x4_f32(false, a, false, b, (short)0, c,
                                               false, false);
}

// Max-reduce across each 16-lane row via DPP butterfly (VALU, no LDS traffic).
__device__ __forceinline__ float row_max16(float x) {
  x = fmaxf(x, __int_as_float(__builtin_amdgcn_update_dpp(
          0, __float_as_int(x), 0xB1 /*quad_perm 1,0,3,2*/, 0xF, 0xF, true)));
  x = fmaxf(x, __int_as_float(__builtin_amdgcn_update_dpp(
          0, __float_as_int(x), 0x4E /*quad_perm 2,3,0,1*/, 0xF, 0xF, true)));
  x = fmaxf(x, __int_as_float(__builtin_amdgcn_update_dpp(
          0, __float_as_int(x), 0x141 /*row_half_mirror*/, 0xF, 0xF, true)));
  x = fmaxf(x, __int_as_float(__builtin_amdgcn_update_dpp(
          0, __float_as_int(x), 0x140 /*row_mirror*/, 0xF, 0xF, true)));
  return x;
}

// Async DMA: global -> LDS, 16 bytes per lane, tracked by ASYNCcnt.
__device__ __forceinline__ void async_load_b128(unsigned lds_off, const float* gaddr) {
  asm volatile("global_load_async_to_lds_b128 %0, %1, off"
               :: "v"(lds_off), "v"(gaddr) : "memory");
}

__device__ __forceinline__ unsigned lds_off(const void* p) {
  // LDS aperture: flat addr[31:0] is the wave-relative LDS byte offset.
  return (unsigned)(size_t)p;
}

__global__ __launch_bounds__(128, 2)
void radix_attn_extend(const float* __restrict__ q,
                       const float* __restrict__ kn,
                       const float* __restrict__ vn,
                       const float* __restrict__ pk,
                       const float* __restrict__ pv,
                       float* __restrict__ out)
{
  __shared__ float kt[2][16 * RSTR];   // K tiles (double buffered), [key][d]
  __shared__ float vt[2][16 * RSTR];   // V tiles (double buffered), [key][d]
  __shared__ float st[4][16 * SSTR];   // per-wave P staging [query][key]

  const int MT  = E_ / 16;
  const int bid = blockIdx.x;
  const int b   = bid / (HKV_ * MT);
  const int rem = bid % (HKV_ * MT);
  const int h   = rem / MT;            // kv head
  const int mt  = rem % MT;
  const int m0  = mt * 16;             // first query of this tile

  const int tid  = threadIdx.x;
  const int wave = tid >> 5;           // query head within group: g = wave
  const int lane = tid & 31;
  const int half = lane >> 4;
  const int l16  = lane & 15;
  const int hq   = h * G_ + wave;      // global query head

  // ---- Load Q tile into A-fragments (pre-scaled) ----
  v2f qf[32];
  {
    const float* qrow = q + (size_t)(b * E_ + m0 + l16) * (HQ_ * D_) + hq * D_;
    #pragma unroll
    for (int kc = 0; kc < 32; ++kc) {
      const int c = kc * 4 + half * 2;
      v2f t;
      t.x = qrow[c] * SCALE_;
      t.y = qrow[c + 1] * SCALE_;
      qf[kc] = t;
    }
  }

  // Output accumulators (8 d-tiles) + row-sum tile (ones-column trick)
  v8f acc[8];
  #pragma unroll
  for (int t = 0; t < 8; ++t) { v8f z = {}; acc[t] = z; }
  v8f accR = {};
  float rm[8];
  #pragma unroll
  for (int r = 0; r < 8; ++r) rm[r] = -INFINITY;

  const v2f ones = {1.0f, 1.0f};

  const int NP = P_ / 16;        // prefix key blocks (non-causal)
  const int NC = mt + 1;         // new-token key blocks (causal)
  const int NB = NP + NC;

  // Cooperative loader role: this thread owns 64B of K and V per tile.
  const int key = tid >> 3;           // 0..15
  const int c0  = (tid & 7) * 16;     // 0..112

  const unsigned kOff[2] = { lds_off(&kt[0][key * RSTR + c0]),
                             lds_off(&kt[1][key * RSTR + c0]) };
  const unsigned vOff[2] = { lds_off(&vt[0][key * RSTR + c0]),
                             lds_off(&vt[1][key * RSTR + c0]) };

  // Issue 8 async b128 DMAs (4 K + 4 V) for key block bb into buffer bf.
  auto issue_tiles = [&](int bb, int bf) {
    const float *ksrc, *vsrc;
    if (bb < NP) {
      const size_t base = ((size_t)(b * P_ + bb * 16 + key) * HKV_ + h) * D_ + c0;
      ksrc = pk + base; vsrc = pv + base;
    } else {
      const int k0 = (bb - NP) * 16;
      const size_t base = (size_t)(b * E_ + k0 + key) * (HKV_ * D_) + h * D_ + c0;
      ksrc = kn + base; vsrc = vn + base;
    }
    #pragma unroll
    for (int j = 0; j < 4; ++j) {
      async_load_b128(kOff[bf] + j * 16, ksrc + j * 4);
      async_load_b128(vOff[bf] + j * 16, vsrc + j * 4);
    }
  };

  // ---- Software pipeline: prime buffer 0 ----
  issue_tiles(0, 0);
  int buf = 0;

  for (int blk = 0; blk < NB; ++blk) {
    const bool isPrefix = blk < NP;
    const int  kb0      = isPrefix ? blk * 16 : (blk - NP) * 16;

    // Issue next block into the other buffer, then wait for current block's 8
    // transfers (async loads complete in order -> waiting <=8 releases them
    // while the next 8 stay in flight).
    if (blk + 1 < NB) {
      issue_tiles(blk + 1, buf ^ 1);
      asm volatile("s_wait_asynccnt 0x8" ::: "memory");
    } else {
      asm volatile("s_wait_asynccnt 0x0" ::: "memory");
    }
    __syncthreads();   // all waves' DMAs for this buffer have landed

    const float* kb = &kt[buf][0];
    const float* vb = &vt[buf][0];

    // ---- S = Q * K^T  (32 chained WMMAs over D) ----
    v8f s = {};
    #pragma unroll
    for (int kc = 0; kc < 32; ++kc) {
      const int c = kc * 4 + half * 2;
      v2f bf2 = *(const v2f*)&kb[l16 * RSTR + c];
      s = wmma4_f32(qf[kc], bf2, s);
    }

    // ---- Causal mask (only last new-token block; uniform branch) ----
    if (!isPrefix && kb0 == m0) {
      #pragma unroll
      for (int r = 0; r < 8; ++r) {
        const int M = r + 8 * half;
        if (l16 > M) s[r] = -INFINITY;
      }
    }

    // ---- Online softmax: row max via DPP, probabilities via exp ----
    float alpha[8];
    #pragma unroll
    for (int r = 0; r < 8; ++r) {
      const float cur = row_max16(s[r]);
      const float nm  = fmaxf(rm[r], cur);
      alpha[r] = __expf(rm[r] - nm);
      rm[r] = nm;
      s[r] = __expf(s[r] - nm);
    }

    // ---- Rescale accumulators (incl. row-sum tile) ----
    #pragma unroll
    for (int r = 0; r < 8; ++r) {
      accR[r] *= alpha[r];
      #pragma unroll
      for (int t = 0; t < 8; ++t)
        acc[t][r] *= alpha[r];
    }

    // ---- P: C-layout -> A-layout via per-wave LDS staging ----
    float* sw = &st[wave][0];
    #pragma unroll
    for (int r = 0; r < 8; ++r)
      sw[(r + 8 * half) * SSTR + l16] = s[r];
    asm volatile("s_wait_dscnt 0" ::: "memory");   // intra-wave transpose fence

    // ---- O += P * V ; row-sum += P * 1  (32 + 4 WMMAs) ----
    #pragma unroll
    for (int kc = 0; kc < 4; ++kc) {
      const int c = kc * 4 + 2 * half;             // key row pair in V tile
      v2f pa = *(const v2f*)&sw[l16 * SSTR + c];
      accR = wmma4_f32(pa, ones, accR);            // running row-sum
      #pragma unroll
      for (int t = 0; t < 8; ++t) {
        v2f bv;                                    // B: K=key rows c,c+1; N=d col
        bv.x = vb[(c + 0) * RSTR + t * 16 + l16];
        bv.y = vb[(c + 1) * RSTR + t * 16 + l16];
        acc[t] = wmma4_f32(pa, bv, acc[t]);
      }
    }
    __syncthreads();   // everyone done with this buffer before it is re-filled
    buf ^= 1;
  }

  // ---- Normalize and write output [B*E, HQ*D] ----
  #pragma unroll
  for (int r = 0; r < 8; ++r) {
    const float invr = 1.0f / accR[r];             // row-sum from ones-column tile
    const int M = r + 8 * half;
    float* orow = out + (size_t)(b * E_ + m0 + M) * (HQ_ * D_) + hq * D_ + l16;
    #pragma unroll
    for (int t = 0; t < 8; ++t)
      orow[t * 16] = acc[t][r] * invr;
  }
}

extern "C" void kernel_launch(void* const* d_in, const int* in_sizes, int n_in,
                              void* d_out, int out_size, void* d_ws, size_t ws_size,
                              hipStream_t stream) {
  const float* q  = (const float*)d_in[0];
  const float* k  = (const float*)d_in[1];
  const float* v  = (const float*)d_in[2];
  const float* pk = (const float*)d_in[3];
  const float* pv = (const float*)d_in[4];
  float* out = (float*)d_out;

  const int grid = B_ * HKV_ * (E_ / 16);   // 1024 workgroups
  radix_attn_extend<<<grid, 128, 0, stream>>>(q, k, v, pk, pv, out);
}